// GNN_18528488915063
// MI455X (gfx1250) — compile-verified
//
#include <hip/hip_runtime.h>

typedef __attribute__((ext_vector_type(2))) float    v2f;
typedef __attribute__((ext_vector_type(4))) float    v4f;
typedef __attribute__((ext_vector_type(8))) float    v8f;
typedef __attribute__((ext_vector_type(4))) unsigned v4u;
typedef __attribute__((ext_vector_type(8))) unsigned v8u;

#define N_NODES  100000
#define N_EDGES  640000
#define N_GRAPHS 64
#define HIDDEN   128

// ---------------------------------------------------------------- utilities
__global__ void gnn_zero_kernel(float* __restrict__ p, int n) {
  int i = blockIdx.x * blockDim.x + threadIdx.x;
  int stride = gridDim.x * blockDim.x;
  for (; i < n; i += stride) p[i] = 0.0f;
}

// ------------------------------------------------- layer 1 (IN_DIM = 2) path
__global__ void gnn_scatter2_kernel(const float* __restrict__ x,
                                    const int* __restrict__ src,
                                    const int* __restrict__ dst,
                                    float* __restrict__ agg2) {
  int e = blockIdx.x * blockDim.x + threadIdx.x;
  if (e >= N_EDGES) return;
  int s = src[e], d = dst[e];
  atomicAdd(&agg2[d * 2 + 0], x[s * 2 + 0]);
  atomicAdd(&agg2[d * 2 + 1], x[s * 2 + 1]);
}

__global__ void gnn_layer1_kernel(const float* __restrict__ x,
                                  const float* __restrict__ agg2,
                                  const float* __restrict__ Wr,   // [2,128]
                                  const float* __restrict__ br,   // [128]
                                  const float* __restrict__ Wt,   // [2,128]
                                  float* __restrict__ h) {
  int idx = blockIdx.x * blockDim.x + threadIdx.x;
  if (idx >= N_NODES * HIDDEN) return;
  int i = idx >> 7, c = idx & 127;
  float v = agg2[i * 2 + 0] * Wr[c] + agg2[i * 2 + 1] * Wr[HIDDEN + c] + br[c]
          + x[i * 2 + 0] * Wt[c] + x[i * 2 + 1] * Wt[HIDDEN + c];
  h[idx] = fmaxf(v, 0.0f);
}

// ----------------------------------------- 128-wide edge scatter-add (L2 atomics)
__global__ void gnn_scatter128_kernel(const float* __restrict__ h,
                                      const int* __restrict__ src,
                                      const int* __restrict__ dst,
                                      float* __restrict__ agg) {
  int idx = blockIdx.x * blockDim.x + threadIdx.x;   // < 640000*128 = 81.92M
  if (idx >= N_EDGES * HIDDEN) return;
  int e = idx >> 7, c = idx & 127;
  atomicAdd(&agg[dst[e] * HIDDEN + c], h[src[e] * HIDDEN + c]);
}

// --------------------------------------------------- TDM: 128x128 fp32 -> LDS
// Builds a Tensor DMA Descriptor (D#) for a dense 2D tile (tile == tensor,
// 128x128 fp32, stride 128) and issues TENSOR_LOAD_TO_LDS (VADDR2/3 = NULL,
// 2D form). Issued once per workgroup by wave 0; TDM ignores EXEC and is
// tracked with TENSORcnt. Inline asm is used (portable across toolchains).
__device__ __forceinline__ void tdm_load_w128(const float* __restrict__ gsrc,
                                              float* lds_dst) {
  unsigned long long ga = (unsigned long long)(uintptr_t)gsrc;
  // low 32 bits of an LDS generic pointer == LDS byte offset
  unsigned lds = (unsigned)(uintptr_t)lds_dst;

  v4u g0;
  g0[0] = 1u;                                            // count=1, user D#
  g0[1] = lds;                                           // lds_addr
  g0[2] = (unsigned)(ga & 0xFFFFFFFFull);                // global_addr[31:0]
  g0[3] = (unsigned)((ga >> 32) & 0x01FFFFFFull)         // global_addr[56:32]
        | (2u << 30);                                    // type = 2 ("image")

  v8u g1;
  g1[0] = (2u << 16);      // workgroup_mask=0, data_size=2 (4 bytes), no flags
  g1[1] = (128u << 16);    // tensor_dim0[15:0] = 128 (atomic_barrier_addr = 0)
  g1[2] = (128u << 16);    // tensor_dim1[15:0] = 128 (tensor_dim0 hi = 0)
  g1[3] = (128u << 16);    // tile_dim0 = 128     (tensor_dim1 hi = 0)
  g1[4] = 128u;            // tile_dim1 = 128, tile_dim2 = 0 (2D)
  g1[5] = 128u;            // tensor_dim0_stride[31:0] = 128 elements
  g1[6] = 0u;              // stride0 hi = 0, tensor_dim1_stride lo = 0 (unused)
  g1[7] = 0u;              // tensor_dim1_stride hi = 0

  asm volatile("tensor_load_to_lds %0, %1"
               :
               : "s"(g0), "s"(g1)
               : "memory");
}

// ------------------------------------------------------------- fused WMMA GEMM
// H <- (do_relu? relu : id)( A @ Wrel + brel + H @ Wroot ), done IN PLACE.
// Each wave owns 16 rows, holds eight 16x16 f32 accumulators (all N=128),
// and uses V_WMMA_F32_16X16X4_F32 (exact fp32 matrix op, wave32).
// Weights are staged into LDS by the Tensor Data Mover (2 descriptors).
__global__ __launch_bounds__(256) void gnn_gemm_wmma_kernel(
    const float* __restrict__ A,      // agg  [N_NODES,128]
    float* __restrict__ H,            // h    [N_NODES,128], in/out
    const float* __restrict__ Wrel,   // [128,128] row-major (K,N)
    const float* __restrict__ brel,   // [128]
    const float* __restrict__ Wroot,  // [128,128]
    int do_relu) {
  extern __shared__ float smem[];                 // 2 * 128*128 floats = 128 KB
  float* sWrel  = smem;
  float* sWroot = smem + HIDDEN * HIDDEN;

  // wave 0 issues both TDM tile loads, waits on TENSORcnt, then the
  // workgroup barrier publishes the LDS contents to all 8 waves.
  if (threadIdx.x < 32) {
    tdm_load_w128(Wrel,  sWrel);
    tdm_load_w128(Wroot, sWroot);
    __builtin_amdgcn_s_wait_tensorcnt(0);
  }
  __syncthreads();

  const int wave = threadIdx.x >> 5;
  const int lane = threadIdx.x & 31;
  const int rowBase = (blockIdx.x * 8 + wave) * 16;
  if (rowBase >= N_NODES) return;                 // uniform per wave, EXEC stays full

  const int m    = lane & 15;
  const int half = lane >> 4;                     // 0 or 1
  const int koff = half * 2;
  const float* aRow = A + (rowBase + m) * HIDDEN;
  const float* hRow = H + (rowBase + m) * HIDDEN;

  v8f acc[8];
  #pragma unroll
  for (int nt = 0; nt < 8; ++nt) {                // init C with bias: C[m][n] = brel[n]
    float b = brel[nt * 16 + m];
    #pragma unroll
    for (int v = 0; v < 8; ++v) acc[nt][v] = b;
  }

  #pragma unroll 2
  for (int k = 0; k < HIDDEN; k += 4) {
    v2f aA = *(const v2f*)(aRow + k + koff);
    v2f aH = *(const v2f*)(hRow + k + koff);
    #pragma unroll
    for (int nt = 0; nt < 8; ++nt) {
      int n = nt * 16 + m;
      v2f bR, bT;
      bR.x = sWrel [(k + koff)     * HIDDEN + n];
      bR.y = sWrel [(k + koff + 1) * HIDDEN + n];
      bT.x = sWroot[(k + koff)     * HIDDEN + n];
      bT.y = sWroot[(k + koff + 1) * HIDDEN + n];
      acc[nt] = __builtin_amdgcn_wmma_f32_16x16x4_f32(
          false, aA, false, bR, (short)0, acc[nt], false, false);
      acc[nt] = __builtin_amdgcn_wmma_f32_16x16x4_f32(
          false, aH, false, bT, (short)0, acc[nt], false, false);
    }
  }

  // D layout: VGPR v -> M = v + 8*half, N = nt*16 + (lane&15). All reads of this
  // wave's rows are complete, so in-place writeback to H is safe.
  #pragma unroll
  for (int nt = 0; nt < 8; ++nt) {
    int n = nt * 16 + m;
    #pragma unroll
    for (int v = 0; v < 8; ++v) {
      float val = acc[nt][v];
      if (do_relu) val = fmaxf(val, 0.0f);
      H[(rowBase + v + half * 8) * HIDDEN + n] = val;
    }
  }
}

// ------------------------------------------------------------------- pooling
__global__ void gnn_pool_kernel(const float* __restrict__ h,
                                const int* __restrict__ batch,
                                float* __restrict__ pooled,
                                float* __restrict__ counts) {
  int idx = blockIdx.x * blockDim.x + threadIdx.x;
  if (idx >= N_NODES * HIDDEN) return;
  int i = idx >> 7, c = idx & 127;
  int g = batch[i];
  atomicAdd(&pooled[g * HIDDEN + c], h[idx]);
  if (c == 0) atomicAdd(&counts[g], 1.0f);
}

__global__ void gnn_head_kernel(const float* __restrict__ pooled,
                                const float* __restrict__ counts,
                                const float* __restrict__ Wimp,  // [128,3]
                                const float* __restrict__ bimp,  // [3]
                                const float* __restrict__ Wst,   // [131,3]
                                const float* __restrict__ bst,   // [3]
                                float* __restrict__ out) {
  __shared__ float mean[HIDDEN];
  __shared__ float imp[3];
  int g = blockIdx.x;
  int t = threadIdx.x;
  float cnt = fmaxf(counts[g], 1.0f);
  if (t < HIDDEN) mean[t] = pooled[g * HIDDEN + t] / cnt;
  __syncthreads();
  if (t < 3) {
    float s = bimp[t];
    for (int c = 0; c < HIDDEN; ++c) s += mean[c] * Wimp[c * 3 + t];
    imp[t] = s;
    out[g * 3 + t] = s;                              // importance [64,3]
  }
  __syncthreads();
  if (t < 3) {
    float s = bst[t];
    for (int c = 0; c < HIDDEN; ++c) s += mean[c] * Wst[c * 3 + t];
    for (int j = 0; j < 3; ++j)      s += imp[j] * Wst[(HIDDEN + j) * 3 + t];
    out[N_GRAPHS * 3 + g * 3 + t] = s;               // state [64,3]
  }
}

// ------------------------------------------------------------------- launcher
extern "C" void kernel_launch(void* const* d_in, const int* in_sizes, int n_in,
                              void* d_out, int out_size, void* d_ws, size_t ws_size,
                              hipStream_t stream) {
  (void)in_sizes; (void)n_in; (void)out_size; (void)ws_size;

  const float* x       = (const float*)d_in[0];
  const int*   edge    = (const int*)d_in[1];      // [2, E]
  const int*   batch   = (const int*)d_in[2];
  const float* W_rel1  = (const float*)d_in[3];
  const float* b_rel1  = (const float*)d_in[4];
  const float* W_root1 = (const float*)d_in[5];
  const float* W_rel2  = (const float*)d_in[6];
  const float* b_rel2  = (const float*)d_in[7];
  const float* W_root2 = (const float*)d_in[8];
  const float* W_rel3  = (const float*)d_in[9];
  const float* b_rel3  = (const float*)d_in[10];
  const float* W_root3 = (const float*)d_in[11];
  const float* W_imp   = (const float*)d_in[12];
  const float* b_imp   = (const float*)d_in[13];
  const float* W_state = (const float*)d_in[14];
  const float* b_state = (const float*)d_in[15];
  float* out = (float*)d_out;

  const int* src = edge;
  const int* dst = edge + N_EDGES;

  // workspace layout (all offsets 256B aligned)
  char* ws = (char*)d_ws;
  float* agg    = (float*)(ws);                               // 12.8M floats
  float* h      = (float*)(ws + 51200000);                    // 12.8M floats
  float* agg2   = (float*)(ws + 102400000);                   // 200K floats
  float* pooled = (float*)(ws + 103200256);                   // 8192 floats
  float* counts = pooled + N_GRAPHS * HIDDEN;                 // 64 floats (contiguous)

  const int T = 256;
  const int nodeElems = N_NODES * HIDDEN;                     // 12.8M
  const int edgeElems = N_EDGES * HIDDEN;                     // 81.92M
  const size_t gemmLds = 2u * HIDDEN * HIDDEN * sizeof(float); // 128 KB (WGP has 320 KB)
  const int gemmBlocks = (N_NODES / 16 + 7) / 8;              // 782

  // ---- layer 1 (K=2, scalar path) ----
  hipLaunchKernelGGL(gnn_zero_kernel, dim3((N_NODES * 2 + T - 1) / T), dim3(T), 0, stream,
                     agg2, N_NODES * 2);
  hipLaunchKernelGGL(gnn_scatter2_kernel, dim3((N_EDGES + T - 1) / T), dim3(T), 0, stream,
                     x, src, dst, agg2);
  hipLaunchKernelGGL(gnn_layer1_kernel, dim3((nodeElems + T - 1) / T), dim3(T), 0, stream,
                     x, agg2, W_rel1, b_rel1, W_root1, h);

  // ---- layer 2: scatter + fused WMMA GEMM (in-place, ReLU) ----
  hipLaunchKernelGGL(gnn_zero_kernel, dim3((nodeElems + T - 1) / T), dim3(T), 0, stream,
                     agg, nodeElems);
  hipLaunchKernelGGL(gnn_scatter128_kernel, dim3((edgeElems + T - 1) / T), dim3(T), 0, stream,
                     h, src, dst, agg);
  hipLaunchKernelGGL(gnn_gemm_wmma_kernel, dim3(gemmBlocks), dim3(T), gemmLds, stream,
                     agg, h, W_rel2, b_rel2, W_root2, 1);

  // ---- layer 3: scatter + fused WMMA GEMM (in-place, no ReLU) ----
  hipLaunchKernelGGL(gnn_zero_kernel, dim3((nodeElems + T - 1) / T), dim3(T), 0, stream,
                     agg, nodeElems);
  hipLaunchKernelGGL(gnn_scatter128_kernel, dim3((edgeElems + T - 1) / T), dim3(T), 0, stream,
                     h, src, dst, agg);
  hipLaunchKernelGGL(gnn_gemm_wmma_kernel, dim3(gemmBlocks), dim3(T), gemmLds, stream,
                     agg, h, W_rel3, b_rel3, W_root3, 0);

  // ---- global mean pool + heads ----
  hipLaunchKernelGGL(gnn_zero_kernel, dim3((N_GRAPHS * HIDDEN + 64 + T - 1) / T), dim3(T), 0, stream,
                     pooled, N_GRAPHS * HIDDEN + N_GRAPHS);
  hipLaunchKernelGGL(gnn_pool_kernel, dim3((nodeElems + T - 1) / T), dim3(T), 0, stream,
                     h, batch, pooled, counts);
  hipLaunchKernelGGL(gnn_head_kernel, dim3(N_GRAPHS), dim3(HIDDEN), 0, stream,
                     pooled, counts, W_imp, b_imp, W_state, b_state, out);
}